// Subsampling_Layer_90915867722284
// MI455X (gfx1250) — compile-verified
//
#include <hip/hip_runtime.h>
#include <math.h>

#define RES 256
#define DEC 8
#define NUM_MEAS (RES * RES / DEC) /* 8192 */
#define BATCH 4

typedef __attribute__((ext_vector_type(2))) float v2f;
typedef __attribute__((ext_vector_type(8))) float v8f;

// ---------------------------------------------------------------------------
// Kernel 1: bilinear gather (grid_sample style, align_corners=True, zero pad)
// input: [B,1,H,W,2] fp32; coords: [M,2]; sC: [B*M] float2 (re,im)
// ---------------------------------------------------------------------------
__global__ void gather_kernel(const float* __restrict__ img,
                              const float* __restrict__ coords,
                              float2* __restrict__ sC) {
  int idx = blockIdx.x * blockDim.x + threadIdx.x;
  if (idx >= BATCH * NUM_MEAS) return;
  int b = idx / NUM_MEAS;
  int m = idx - b * NUM_MEAS;

  float cx = coords[2 * m + 0];
  float cy = coords[2 * m + 1];
  float gx = (cx * (2.0f / RES) + 1.0f) * 0.5f * (float)(RES - 1);
  float gy = (cy * (2.0f / RES) + 1.0f) * 0.5f * (float)(RES - 1);
  float x0f = floorf(gx), y0f = floorf(gy);
  float wx1 = gx - x0f, wx0 = 1.0f - wx1;
  float wy1 = gy - y0f, wy0 = 1.0f - wy1;
  int x0 = (int)x0f, y0 = (int)y0f;

  float re = 0.0f, im = 0.0f;
#pragma unroll
  for (int dy = 0; dy < 2; ++dy) {
#pragma unroll
    for (int dx = 0; dx < 2; ++dx) {
      int xi = x0 + dx, yi = y0 + dy;
      float w = (dx ? wx1 : wx0) * (dy ? wy1 : wy0);
      bool valid = (xi >= 0) && (xi < RES) && (yi >= 0) && (yi < RES);
      int xc = min(max(xi, 0), RES - 1);
      int yc = min(max(yi, 0), RES - 1);
      const float* p = img + ((((size_t)b * RES + yc) * RES + xc) * 2);
      float wv = valid ? w : 0.0f;
      re = fmaf(p[0], wv, re);
      im = fmaf(p[1], wv, im);
    }
  }
  sC[idx] = make_float2(re, im);
}

// ---------------------------------------------------------------------------
// Kernel 2: phase tables.
//   ExC[m][x] = exp(2pi*i * kx_m * (x - N/2) / N)   (row-major [m][x], float2)
//   EyC[y][m] = exp(2pi*i * ky_m * (y - N/2) / N)   ([y][m] so K is contiguous)
// ---------------------------------------------------------------------------
__global__ void phase_kernel(const float* __restrict__ coords,
                             float2* __restrict__ ExC,
                             float2* __restrict__ EyC) {
  int idx = blockIdx.x * blockDim.x + threadIdx.x;
  if (idx >= NUM_MEAS * RES) return;
  int m = idx / RES;
  int n = idx - m * RES;
  float fn = (float)n - (float)(RES / 2);
  const float w = 6.2831853071795864769f / (float)RES;
  float kx = coords[2 * m + 0];
  float ky = coords[2 * m + 1];
  float sx, cx;
  sincosf(w * kx * fn, &sx, &cx);
  ExC[(size_t)m * RES + n] = make_float2(cx, sx);
  float sy, cy;
  sincosf(w * ky * fn, &sy, &cy);
  EyC[(size_t)n * NUM_MEAS + m] = make_float2(cy, sy);
}

// ---------------------------------------------------------------------------
// Kernel 3: complex GEMM via V_WMMA_F32_16X16X4_F32, 4 independent
// accumulators so consecutive WMMAs never chain on the same C matrix:
//   P0 = sum aRe*bRe   P1 = sum aIm*bIm   P2 = sum aRe*bIm   P3 = sum aIm*bRe
//   Re = P0 - P1 ;  Im = P2 + P3   (combined once at epilogue)
// Per wave: one 16(y) x 16(x) output tile of one batch; K = 8192 in steps of 4.
// ---------------------------------------------------------------------------
__global__ void __launch_bounds__(256) nufft_wmma_kernel(
    const float2* __restrict__ sC, const float2* __restrict__ ExC,
    const float2* __restrict__ EyC, float* __restrict__ out) {
  int wave = blockIdx.x * (blockDim.x >> 5) + (threadIdx.x >> 5);
  int lane = threadIdx.x & 31;

  int b = wave >> 8;          // 4 batches
  int ty = (wave >> 4) & 15;  // y tile
  int tx = wave & 15;         // x tile

  int row = lane & 15;           // A row (y_local) and B col (x_local)
  int khalf = (lane >> 4) << 1;  // lanes 0-15 hold K={0,1}, 16-31 hold K={2,3}

  int y = ty * 16 + row;  // A-matrix row this lane feeds
  int x = tx * 16 + row;  // B-matrix column this lane feeds

  const float2* __restrict__ sRow = sC + (size_t)b * NUM_MEAS;
  const float2* __restrict__ eyRow = EyC + (size_t)y * NUM_MEAS;

  v8f p0 = {};
  v8f p1 = {};
  v8f p2 = {};
  v8f p3 = {};

  for (int m0 = 0; m0 < NUM_MEAS; m0 += 4) {
    int mA = m0 + khalf;

    // ---- A fragments: t = s * Ey (complex), two K values per lane ----
    float2 s0 = sRow[mA];
    float2 s1 = sRow[mA + 1];
    float2 e0 = eyRow[mA];  // contiguous pair -> merged b128 load
    float2 e1 = eyRow[mA + 1];

    v2f aRe, aIm;
    aRe.x = fmaf(s0.x, e0.x, -s0.y * e0.y);
    aIm.x = fmaf(s0.x, e0.y, s0.y * e0.x);
    aRe.y = fmaf(s1.x, e1.x, -s1.y * e1.y);
    aIm.y = fmaf(s1.x, e1.y, s1.y * e1.x);

    // ---- B fragments: Ex rows (coalesced across lanes 0-15 in x) ----
    float2 x0 = ExC[(size_t)mA * RES + x];
    float2 x1 = ExC[(size_t)(mA + 1) * RES + x];
    v2f bRe, bIm;
    bRe.x = x0.x;
    bRe.y = x1.x;
    bIm.x = x0.y;
    bIm.y = x1.y;

    // 4 WMMAs per complex K=4 step, each to its own accumulator
    p0 = __builtin_amdgcn_wmma_f32_16x16x4_f32(false, aRe, false, bRe,
                                               (short)0, p0, false, false);
    p1 = __builtin_amdgcn_wmma_f32_16x16x4_f32(false, aIm, false, bIm,
                                               (short)0, p1, false, false);
    p2 = __builtin_amdgcn_wmma_f32_16x16x4_f32(false, aRe, false, bIm,
                                               (short)0, p2, false, false);
    p3 = __builtin_amdgcn_wmma_f32_16x16x4_f32(false, aIm, false, bRe,
                                               (short)0, p3, false, false);
  }

  // ---- combine + store: D VGPR r -> (M = r + 8*(lane>=16), N = lane&15) ----
  int colx = tx * 16 + (lane & 15);
  int rowBase = ty * 16 + ((lane >> 4) << 3);
  float* outB = out + (size_t)b * RES * RES * 2;
#pragma unroll
  for (int r = 0; r < 8; ++r) {
    int yy = rowBase + r;
    float2 v = make_float2(p0[r] - p1[r], p2[r] + p3[r]);
    *(float2*)(outB + (((size_t)yy * RES + colx) * 2)) = v;
  }
}

// ---------------------------------------------------------------------------
extern "C" void kernel_launch(void* const* d_in, const int* in_sizes, int n_in,
                              void* d_out, int out_size, void* d_ws,
                              size_t ws_size, hipStream_t stream) {
  const float* input = (const float*)d_in[0];   // [4,1,256,256,2]
  const float* coords = (const float*)d_in[1];  // [8192,2]

  char* ws = (char*)d_ws;
  float2* sC = (float2*)ws;  // 256 KB
  float2* ExC = (float2*)(ws + sizeof(float2) * (size_t)BATCH * NUM_MEAS);
  float2* EyC = (float2*)(ws + sizeof(float2) * ((size_t)BATCH * NUM_MEAS +
                                                 (size_t)NUM_MEAS * RES));

  gather_kernel<<<(BATCH * NUM_MEAS + 255) / 256, 256, 0, stream>>>(
      input, coords, sC);
  phase_kernel<<<(NUM_MEAS * RES + 255) / 256, 256, 0, stream>>>(coords, ExC,
                                                                 EyC);
  // 4 batches * 16*16 tiles = 1024 waves; 8 waves (256 thr) per block
  nufft_wmma_kernel<<<(BATCH * 16 * 16) / 8, 256, 0, stream>>>(
      sC, ExC, EyC, (float*)d_out);
}